// GAT_6605659702083
// MI455X (gfx1250) — compile-verified
//
#include <hip/hip_runtime.h>
#include <hip/hip_bf16.h>
#include <math.h>

#define N_NODES 20000
#define N_EDGES 320000
#define F_INF   256
#define HEADS   4
#define CHAN    128
#define HID     512
#define GRAPHS  64
#define NCLS    10
#define CLSHID  50
#define NEG_SLOPE 0.2f
#define E_TOT   (N_EDGES + N_NODES)

typedef float v2f __attribute__((ext_vector_type(2)));
typedef float v8f __attribute__((ext_vector_type(8)));

// ---------------------------------------------------------------------------
// GEMM: C[M x Nw] = A[M x K] @ B[K x Nw], all row-major f32, Nw == 512.
// Each wave computes a 16(M) x 64(N) strip = 4 WMMA tiles, sharing one A
// fragment per k-step across 4 V_WMMA_F32_16X16X4_F32 issues. Fragments are
// software double-buffered: next k-step's loads are issued before the current
// WMMAs so the pre-WMMA waits keep 9 loads in flight instead of draining.
// Block = 8 waves covering one 16-row stripe across all 512 columns.
// A layout per lane: row = tile_m*16 + lane%16, K = k0 + 2*(lane/16) + {0,1}
// B layout per lane: col = base + lane%16,      K = k0 + 2*(lane/16) + {0,1}
// C/D layout: VGPR r -> row tile_m*16 + r + 8*(lane/16)
// ---------------------------------------------------------------------------
__global__ __launch_bounds__(256) void gemm_wmma_f32(
    const float* __restrict__ A, const float* __restrict__ B,
    float* __restrict__ C, int K, int Nw) {
  const int lane = threadIdx.x & 31;
  const int wave = threadIdx.x >> 5;
  const int half = lane >> 4;
  const int lm   = lane & 15;
  const int tile_m = blockIdx.x;
  const int col0 = wave * 64 + lm;  // 8 waves x 64 cols = 512

  const float* Ap = A + ((long)tile_m * 16 + lm) * K + 2 * half;
  const float* Bp = B + (long)(2 * half) * Nw + col0;
  const long bstep = 4 * (long)Nw;

  v8f acc[4];
#pragma unroll
  for (int t = 0; t < 4; ++t)
    acc[t] = (v8f){0.f, 0.f, 0.f, 0.f, 0.f, 0.f, 0.f, 0.f};

  // prologue: load k-step 0 fragments
  v2f a_cur, b_cur[4];
  a_cur[0] = Ap[0];
  a_cur[1] = Ap[1];
#pragma unroll
  for (int t = 0; t < 4; ++t) {
    b_cur[t][0] = Bp[t * 16];
    b_cur[t][1] = Bp[t * 16 + Nw];
  }

#pragma unroll 2
  for (int k0 = 0; k0 < K - 4; k0 += 4) {
    Ap += 4;
    Bp += bstep;
    // issue next k-step's loads before consuming current fragments
    v2f a_nxt, b_nxt[4];
    a_nxt[0] = Ap[0];
    a_nxt[1] = Ap[1];
#pragma unroll
    for (int t = 0; t < 4; ++t) {
      b_nxt[t][0] = Bp[t * 16];
      b_nxt[t][1] = Bp[t * 16 + Nw];
    }
#pragma unroll
    for (int t = 0; t < 4; ++t) {
      acc[t] = __builtin_amdgcn_wmma_f32_16x16x4_f32(
          /*neg_a=*/false, a_cur, /*neg_b=*/false, b_cur[t],
          /*c_mod=*/(short)0, acc[t], /*reuse_a=*/false, /*reuse_b=*/false);
    }
    a_cur = a_nxt;
#pragma unroll
    for (int t = 0; t < 4; ++t) b_cur[t] = b_nxt[t];
  }
  // epilogue: last k-step
#pragma unroll
  for (int t = 0; t < 4; ++t) {
    acc[t] = __builtin_amdgcn_wmma_f32_16x16x4_f32(
        /*neg_a=*/false, a_cur, /*neg_b=*/false, b_cur[t],
        /*c_mod=*/(short)0, acc[t], /*reuse_a=*/false, /*reuse_b=*/false);
  }

  float* Cp = C + ((long)tile_m * 16 + 8 * half) * Nw + col0;
#pragma unroll
  for (int t = 0; t < 4; ++t)
#pragma unroll
    for (int r = 0; r < 8; ++r) Cp[(long)r * Nw + t * 16] = acc[t][r];
}

// ---------------------------------------------------------------------------
// alpha_src[n,h], alpha_dst[n,h] = sum_c h[n,h,c] * att_{src,dst}[h,c]
// block = 128 threads = 4 waves, one head per wave, one node per block.
// ---------------------------------------------------------------------------
__global__ __launch_bounds__(128) void attn_coef(
    const float* __restrict__ h, const float* __restrict__ att_src,
    const float* __restrict__ att_dst, float* __restrict__ as,
    float* __restrict__ ad) {
  const int n = blockIdx.x;
  const int head = threadIdx.x >> 5;
  const int lane = threadIdx.x & 31;
  const float* hp = h + (long)n * HID + head * CHAN + lane * 4;
  const float* sp = att_src + head * CHAN + lane * 4;
  const float* dp = att_dst + head * CHAN + lane * 4;
  float ssum = 0.f, dsum = 0.f;
#pragma unroll
  for (int i = 0; i < 4; ++i) {
    float v = hp[i];
    ssum += v * sp[i];
    dsum += v * dp[i];
  }
#pragma unroll
  for (int off = 16; off > 0; off >>= 1) {
    ssum += __shfl_xor(ssum, off, 32);
    dsum += __shfl_xor(dsum, off, 32);
  }
  if (lane == 0) {
    as[n * HEADS + head] = ssum;
    ad[n * HEADS + head] = dsum;
  }
}

__global__ void fill_f32(float* __restrict__ p, long n, float v) {
  long i = (long)blockIdx.x * blockDim.x + threadIdx.x;
  long stride = (long)gridDim.x * blockDim.x;
  for (; i < n; i += stride) p[i] = v;
}

__device__ __forceinline__ float leaky(float v) {
  return v > 0.f ? v : NEG_SLOPE * v;
}

__device__ __forceinline__ void atomicMaxF(float* addr, float v) {
  if (v >= 0.f)
    atomicMax((int*)addr, __float_as_int(v));
  else
    atomicMin((unsigned int*)addr, __float_as_uint(v));
}

__device__ __forceinline__ void edge_sd(long e, const int* __restrict__ srcv,
                                        const int* __restrict__ dstv, int& s,
                                        int& d) {
  if (e < N_EDGES) {
    s = srcv[e];
    d = dstv[e];
  } else {
    s = d = (int)(e - N_EDGES);  // appended self loops
  }
}

// pass 1: segment max over destination
__global__ void edge_max(const int* __restrict__ srcv,
                         const int* __restrict__ dstv,
                         const float* __restrict__ as,
                         const float* __restrict__ ad, float* __restrict__ m) {
  long e = (long)blockIdx.x * blockDim.x + threadIdx.x;
  if (e >= E_TOT) return;
  int s, d;
  edge_sd(e, srcv, dstv, s, d);
#pragma unroll
  for (int hh = 0; hh < HEADS; ++hh) {
    float v = leaky(as[s * HEADS + hh] + ad[d * HEADS + hh]);
    atomicMaxF(&m[d * HEADS + hh], v);
  }
}

// pass 2: segment sum of exp(e - m[dst])
__global__ void edge_sum(const int* __restrict__ srcv,
                         const int* __restrict__ dstv,
                         const float* __restrict__ as,
                         const float* __restrict__ ad,
                         const float* __restrict__ m, float* __restrict__ ssum) {
  long e = (long)blockIdx.x * blockDim.x + threadIdx.x;
  if (e >= E_TOT) return;
  int s, d;
  edge_sd(e, srcv, dstv, s, d);
#pragma unroll
  for (int hh = 0; hh < HEADS; ++hh) {
    float v = leaky(as[s * HEADS + hh] + ad[d * HEADS + hh]);
    atomicAdd(&ssum[d * HEADS + hh], __expf(v - m[d * HEADS + hh]));
  }
}

// pass 3: out[dst] += alpha * h[src] ; one wave per edge, float4 per head lane
__global__ __launch_bounds__(256) void edge_aggr(
    const int* __restrict__ srcv, const int* __restrict__ dstv,
    const float* __restrict__ as, const float* __restrict__ ad,
    const float* __restrict__ m, const float* __restrict__ ssum,
    const float* __restrict__ h, float* __restrict__ out) {
  const int wave = threadIdx.x >> 5;
  const int lane = threadIdx.x & 31;
  long e = (long)blockIdx.x * 8 + wave;
  if (e >= E_TOT) return;
  int s, d;
  edge_sd(e, srcv, dstv, s, d);
#pragma unroll
  for (int hh = 0; hh < HEADS; ++hh) {
    float v = leaky(as[s * HEADS + hh] + ad[d * HEADS + hh]);
    float alpha = __expf(v - m[d * HEADS + hh]) / ssum[d * HEADS + hh];
    const float4 hv =
        *(const float4*)(h + (long)s * HID + hh * CHAN + lane * 4);
    float* op = out + (long)d * HID + hh * CHAN + lane * 4;
    atomicAdd(op + 0, alpha * hv.x);
    atomicAdd(op + 1, alpha * hv.y);
    atomicAdd(op + 2, alpha * hv.z);
    atomicAdd(op + 3, alpha * hv.w);
  }
}

__global__ void bias_elu(const float* __restrict__ x,
                         const float* __restrict__ b, float* __restrict__ y,
                         long n) {
  long i = (long)blockIdx.x * blockDim.x + threadIdx.x;
  if (i >= n) return;
  float v = x[i] + b[i & (HID - 1)];
  y[i] = v > 0.f ? v : (__expf(v) - 1.f);
}

// global mean-pool numerator/denominator (bias1 fused in)
__global__ __launch_bounds__(128) void pool_sum(const float* __restrict__ agg,
                                                const float* __restrict__ bias,
                                                const int* __restrict__ batch,
                                                float* __restrict__ gsum,
                                                float* __restrict__ gcnt) {
  const int n = blockIdx.x;
  const int g = batch[n];
  for (int j = threadIdx.x; j < HID; j += blockDim.x)
    atomicAdd(&gsum[g * HID + j], agg[(long)n * HID + j] + bias[j]);
  if (threadIdx.x == 0) atomicAdd(&gcnt[g], 1.0f);
}

// classifier: Linear(512->50) -> ReLU -> Linear(50->10) -> softmax
__global__ __launch_bounds__(64) void classifier(
    const float* __restrict__ gsum, const float* __restrict__ gcnt,
    const float* __restrict__ W1, const float* __restrict__ b1,
    const float* __restrict__ W2, const float* __restrict__ b2,
    float* __restrict__ out) {
  __shared__ float z[CLSHID];
  __shared__ float logit[NCLS];
  __shared__ float red[2];
  const int g = blockIdx.x;
  const int t = threadIdx.x;
  const float inv = 1.f / fmaxf(gcnt[g], 1.f);
  if (t < CLSHID) {
    float acc = b1[t];
    for (int j = 0; j < HID; ++j)
      acc += gsum[g * HID + j] * inv * W1[j * CLSHID + t];
    z[t] = fmaxf(acc, 0.f);
  }
  __syncthreads();
  if (t < NCLS) {
    float acc = b2[t];
    for (int k = 0; k < CLSHID; ++k) acc += z[k] * W2[k * NCLS + t];
    logit[t] = acc;
  }
  __syncthreads();
  if (t == 0) {
    float mx = logit[0];
    for (int i = 1; i < NCLS; ++i) mx = fmaxf(mx, logit[i]);
    float sm = 0.f;
    for (int i = 0; i < NCLS; ++i) sm += __expf(logit[i] - mx);
    red[0] = mx;
    red[1] = sm;
  }
  __syncthreads();
  if (t < NCLS) out[g * NCLS + t] = __expf(logit[t] - red[0]) / red[1];
}

// ---------------------------------------------------------------------------
extern "C" void kernel_launch(void* const* d_in, const int* in_sizes, int n_in,
                              void* d_out, int out_size, void* d_ws,
                              size_t ws_size, hipStream_t stream) {
  const float* x        = (const float*)d_in[0];
  const int*   ei       = (const int*)d_in[1];
  const int*   batch    = (const int*)d_in[2];
  const float* W0       = (const float*)d_in[3];
  const float* att_src0 = (const float*)d_in[4];
  const float* att_dst0 = (const float*)d_in[5];
  const float* bias0    = (const float*)d_in[6];
  const float* W1       = (const float*)d_in[7];
  const float* att_src1 = (const float*)d_in[8];
  const float* att_dst1 = (const float*)d_in[9];
  const float* bias1    = (const float*)d_in[10];
  const float* clsW1    = (const float*)d_in[11];
  const float* clsb1    = (const float*)d_in[12];
  const float* clsW2    = (const float*)d_in[13];
  const float* clsb2    = (const float*)d_in[14];

  const int* srcv = ei;            // edge_index[0]
  const int* dstv = ei + N_EDGES;  // edge_index[1]

  const long NH = (long)N_NODES * HID;   // 10,240,000 floats
  float* buf0 = (float*)d_ws;            // h0, later x1 (post-ELU)
  float* buf1 = buf0 + NH;               // agg0, later h1
  float* buf2 = buf1 + NH;               // agg1
  float* asA  = buf2 + NH;               // [N, H]
  float* adA  = asA + (long)N_NODES * HEADS;
  float* mA   = adA + (long)N_NODES * HEADS;
  float* sA   = mA + (long)N_NODES * HEADS;
  float* gsum = sA + (long)N_NODES * HEADS;  // [G, HID]
  float* gcnt = gsum + (long)GRAPHS * HID;   // [G]

  const int gemm_blocks = N_NODES / 16;  // 1250 blocks x 8 waves
  const int eb = (E_TOT + 255) / 256;
  const int eaggr_blocks = (E_TOT + 7) / 8;

  // ---- layer 0 ----
  gemm_wmma_f32<<<gemm_blocks, 256, 0, stream>>>(x, W0, buf0, F_INF, HID);
  attn_coef<<<N_NODES, 128, 0, stream>>>(buf0, att_src0, att_dst0, asA, adA);
  fill_f32<<<512, 256, 0, stream>>>(mA, (long)N_NODES * HEADS, -INFINITY);
  fill_f32<<<512, 256, 0, stream>>>(sA, (long)N_NODES * HEADS, 0.f);
  fill_f32<<<2048, 256, 0, stream>>>(buf1, NH, 0.f);
  edge_max<<<eb, 256, 0, stream>>>(srcv, dstv, asA, adA, mA);
  edge_sum<<<eb, 256, 0, stream>>>(srcv, dstv, asA, adA, mA, sA);
  edge_aggr<<<eaggr_blocks, 256, 0, stream>>>(srcv, dstv, asA, adA, mA, sA, buf0, buf1);
  bias_elu<<<(int)((NH + 255) / 256), 256, 0, stream>>>(buf1, bias0, buf0, NH);

  // ---- layer 1 ----
  gemm_wmma_f32<<<gemm_blocks, 256, 0, stream>>>(buf0, W1, buf1, HID, HID);
  attn_coef<<<N_NODES, 128, 0, stream>>>(buf1, att_src1, att_dst1, asA, adA);
  fill_f32<<<512, 256, 0, stream>>>(mA, (long)N_NODES * HEADS, -INFINITY);
  fill_f32<<<512, 256, 0, stream>>>(sA, (long)N_NODES * HEADS, 0.f);
  fill_f32<<<2048, 256, 0, stream>>>(buf2, NH, 0.f);
  edge_max<<<eb, 256, 0, stream>>>(srcv, dstv, asA, adA, mA);
  edge_sum<<<eb, 256, 0, stream>>>(srcv, dstv, asA, adA, mA, sA);
  edge_aggr<<<eaggr_blocks, 256, 0, stream>>>(srcv, dstv, asA, adA, mA, sA, buf1, buf2);

  // ---- pool + classifier ----
  fill_f32<<<64, 256, 0, stream>>>(gsum, (long)GRAPHS * HID, 0.f);
  fill_f32<<<1, 64, 0, stream>>>(gcnt, (long)GRAPHS, 0.f);
  pool_sum<<<N_NODES, 128, 0, stream>>>(buf2, bias1, batch, gsum, gcnt);
  classifier<<<GRAPHS, 64, 0, stream>>>(gsum, gcnt, clsW1, clsb1, clsW2, clsb2,
                                        (float*)d_out);
}